// ZBLRepulsionEnergy_17875653886228
// MI455X (gfx1250) — compile-verified
//
#include <hip/hip_runtime.h>
#include <hip/hip_bf16.h>
#include <stdint.h>

#define ZBL_KEHALF 7.199822675975274f   // 14.399645351950548 / 2
#define TILE_PAIRS 128                  // pairs per wave per pipeline step
#define WAVES_PER_BLOCK 8               // 256 threads, wave32
#define WAVE_LDS_BYTES 4096             // 2 buffers * 4 streams * 512B
#define BUF_BYTES 2048                  // 4 streams * 512B

__device__ __forceinline__ float softplus_f(float x) {
    return __logf(1.0f + __expf(x));
}

// ---------------------------------------------------------------------------
// Async stage of one 128-pair tile (4 streams) into wave-private LDS.
// 4 x global_load_async_to_lds_b128 per lane; tracked on ASYNCcnt.
// th:TH_LOAD_NT — the pair streams are read-once (256 MB total); keep them
// from evicting the hot L2-resident atom table / output accumulator.
// ---------------------------------------------------------------------------
__device__ __forceinline__ void zbl_issue_tile(uint32_t lds_buf, int lane, int tile,
                                               const int* __restrict__ idx_i,
                                               const int* __restrict__ idx_j,
                                               const float* __restrict__ rij,
                                               const float* __restrict__ cut) {
    const int p0 = tile * TILE_PAIRS + lane * 4;
    uint64_t gi = (uint64_t)(uintptr_t)(idx_i + p0);
    uint64_t gj = (uint64_t)(uintptr_t)(idx_j + p0);
    uint64_t gr = (uint64_t)(uintptr_t)(rij + p0);
    uint64_t gc = (uint64_t)(uintptr_t)(cut + p0);
    uint32_t l0 = lds_buf + (uint32_t)(lane * 16);
    asm volatile(
        "global_load_async_to_lds_b128 %0, %4, off th:TH_LOAD_NT\n\t"
        "global_load_async_to_lds_b128 %1, %5, off th:TH_LOAD_NT\n\t"
        "global_load_async_to_lds_b128 %2, %6, off th:TH_LOAD_NT\n\t"
        "global_load_async_to_lds_b128 %3, %7, off th:TH_LOAD_NT"
        :
        : "v"(l0), "v"(l0 + 512u), "v"(l0 + 1024u), "v"(l0 + 1536u),
          "v"(gi), "v"(gj), "v"(gr), "v"(gc)
        : "memory");
}

// ---------------------------------------------------------------------------
// Main pair kernel (hot loop; emitted FIRST so the disasm snippet shows it).
// Each wave double-buffers 128-pair tiles through LDS with the async copy
// engine, overlapping HBM streaming with L2 gathers, 4 v_exp_f32, and a
// hardware global_atomic_add_f32 scatter.
// ---------------------------------------------------------------------------
template <bool CACHED>
__global__ __launch_bounds__(256) void zbl_pairs_kernel(
    const float* __restrict__ Zf, const float* __restrict__ rij,
    const float* __restrict__ cut, const int* __restrict__ idx_i,
    const int* __restrict__ idx_j, const float* __restrict__ ws,
    float* __restrict__ out, int ntiles) {
    __shared__ __align__(16) char smem[WAVES_PER_BLOCK * WAVE_LDS_BYTES];

    const int lane = threadIdx.x & 31;
    const int wave = threadIdx.x >> 5;
    const int wgid = blockIdx.x * WAVES_PER_BLOCK + wave;
    const int wstride = gridDim.x * WAVES_PER_BLOCK;

    const float sp_apow = ws[0];
    const float A1 = ws[2], A2 = ws[3], A3 = ws[4], A4 = ws[5];
    const float C1 = ws[6], C2 = ws[7], C3 = ws[8], C4 = ws[9];
    const float2* atoms = (const float2*)(ws + 16);

    char* wbase = smem + wave * WAVE_LDS_BYTES;
    // Low 32 bits of a generic LDS pointer are the workgroup-relative LDS addr.
    const uint32_t wlds = (uint32_t)(uintptr_t)wbase;

    int t = wgid;
    if (t < ntiles) zbl_issue_tile(wlds, lane, t, idx_i, idx_j, rij, cut);
    int buf = 0;

    while (t < ntiles) {
        const int nt = t + wstride;              // wave-uniform
        if (nt < ntiles) {
            zbl_issue_tile(wlds + (uint32_t)((buf ^ 1) * BUF_BYTES), lane, nt,
                           idx_i, idx_j, rij, cut);
            // current tile's 4 loads complete in order before the next 4
            asm volatile("s_wait_asynccnt 0x4" ::: "memory");
        } else {
            asm volatile("s_wait_asynccnt 0x0" ::: "memory");
        }

        const char* b = wbase + buf * BUF_BYTES;
        int4 ii = *(const int4*)(b + 0 + lane * 16);
        int4 jj = *(const int4*)(b + 512 + lane * 16);
        float4 rr = *(const float4*)(b + 1024 + lane * 16);
        float4 cc = *(const float4*)(b + 1536 + lane * 16);

        const int iv[4] = {ii.x, ii.y, ii.z, ii.w};
        const int jv[4] = {jj.x, jj.y, jj.z, jj.w};
        const float rv[4] = {rr.x, rr.y, rr.z, rr.w};
        const float cv[4] = {cc.x, cc.y, cc.z, cc.w};

#pragma unroll
        for (int k = 0; k < 4; ++k) {
            const int i = iv[k];
            const int j = jv[k];
            float zfi, zfj, zi, zj;
            if (CACHED) {
                float2 ai = atoms[i];
                float2 aj = atoms[j];
                zfi = ai.x; zi = ai.y;
                zfj = aj.x; zj = aj.y;
            } else {
                zfi = Zf[i]; zfj = Zf[j];
                zi = __expf(sp_apow * __logf(zfi));
                zj = __expf(sp_apow * __logf(zfj));
            }
            const float r = rv[k];
            const float x = (zi + zj) * r;       // a * r, with sp(adiv) folded into A_k
            float s = C1 * __expf(-A1 * x);
            s += C2 * __expf(-A2 * x);
            s += C3 * __expf(-A3 * x);
            s += C4 * __expf(-A4 * x);
            const float contrib = s * cv[k] * zfi * zfj * __builtin_amdgcn_rcpf(r);
            unsafeAtomicAdd(&out[i], contrib);   // global_atomic_add_f32 (no return)
        }
        buf ^= 1;
        t = nt;
    }
}

// Force early emission so the disasm snippet shows the hot kernel.
template __global__ void zbl_pairs_kernel<true>(const float*, const float*, const float*,
                                                const int*, const int*, const float*, float*, int);
template __global__ void zbl_pairs_kernel<false>(const float*, const float*, const float*,
                                                 const int*, const int*, const float*, float*, int);

// ---------------------------------------------------------------------------
// Stage 0: fold the 10 softplus'd scalar parameters into derived constants.
// hdr[0]=sp(apow) hdr[1]=sp(adiv) hdr[2..5]=A1..A4=sp(a_k)*sp(adiv)
// hdr[6..9]=C1..C4=KEHALF*sp(c_k)/sum(sp(c))
// ---------------------------------------------------------------------------
__global__ void zbl_consts_kernel(const float* __restrict__ adiv, const float* __restrict__ apow,
                                  const float* __restrict__ c1, const float* __restrict__ c2,
                                  const float* __restrict__ c3, const float* __restrict__ c4,
                                  const float* __restrict__ a1, const float* __restrict__ a2,
                                  const float* __restrict__ a3, const float* __restrict__ a4,
                                  float* __restrict__ hdr) {
    if (threadIdx.x == 0 && blockIdx.x == 0) {
        float sp_apow = softplus_f(apow[0]);
        float sp_adiv = softplus_f(adiv[0]);
        float c1p = softplus_f(c1[0]);
        float c2p = softplus_f(c2[0]);
        float c3p = softplus_f(c3[0]);
        float c4p = softplus_f(c4[0]);
        float inv = 1.0f / (c1p + c2p + c3p + c4p);
        hdr[0] = sp_apow;
        hdr[1] = sp_adiv;
        hdr[2] = softplus_f(a1[0]) * sp_adiv;
        hdr[3] = softplus_f(a2[0]) * sp_adiv;
        hdr[4] = softplus_f(a3[0]) * sp_adiv;
        hdr[5] = softplus_f(a4[0]) * sp_adiv;
        hdr[6] = ZBL_KEHALF * c1p * inv;
        hdr[7] = ZBL_KEHALF * c2p * inv;
        hdr[8] = ZBL_KEHALF * c3p * inv;
        hdr[9] = ZBL_KEHALF * c4p * inv;
    }
}

// ---------------------------------------------------------------------------
// Stage 1: zero the output; precompute per-atom {Zf, Zf^sp(apow)} table (L2
// resident: 4 MB << 192 MB) so the pair kernel does one b64 gather per side.
// ---------------------------------------------------------------------------
__global__ void zbl_atoms_kernel(const float* __restrict__ Zf, float* __restrict__ out,
                                 float* __restrict__ ws, int N, int cache_atoms) {
    const float sp_apow = ws[0];
    float2* atoms = (float2*)(ws + 16);
    for (int i = blockIdx.x * blockDim.x + threadIdx.x; i < N; i += gridDim.x * blockDim.x) {
        out[i] = 0.0f;
        if (cache_atoms) {
            float zf = Zf[i];
            float z = __expf(sp_apow * __logf(zf));   // zf >= 1
            atoms[i] = make_float2(zf, z);
        }
    }
}

// ---------------------------------------------------------------------------
// Scalar tail for pair counts not a multiple of TILE_PAIRS (empty for 16M).
// ---------------------------------------------------------------------------
__global__ void zbl_pairs_tail_kernel(const float* __restrict__ Zf, const float* __restrict__ rij,
                                      const float* __restrict__ cut, const int* __restrict__ idx_i,
                                      const int* __restrict__ idx_j, const float* __restrict__ ws,
                                      float* __restrict__ out, int start, int n_pairs, int cached) {
    const float sp_apow = ws[0];
    const float A1 = ws[2], A2 = ws[3], A3 = ws[4], A4 = ws[5];
    const float C1 = ws[6], C2 = ws[7], C3 = ws[8], C4 = ws[9];
    const float2* atoms = (const float2*)(ws + 16);
    for (int p = start + blockIdx.x * blockDim.x + threadIdx.x; p < n_pairs;
         p += gridDim.x * blockDim.x) {
        const int i = idx_i[p];
        const int j = idx_j[p];
        float zfi, zfj, zi, zj;
        if (cached) {
            float2 ai = atoms[i];
            float2 aj = atoms[j];
            zfi = ai.x; zi = ai.y; zfj = aj.x; zj = aj.y;
        } else {
            zfi = Zf[i]; zfj = Zf[j];
            zi = __expf(sp_apow * __logf(zfi));
            zj = __expf(sp_apow * __logf(zfj));
        }
        const float r = rij[p];
        const float x = (zi + zj) * r;
        float s = C1 * __expf(-A1 * x) + C2 * __expf(-A2 * x) + C3 * __expf(-A3 * x) +
                  C4 * __expf(-A4 * x);
        const float contrib = s * cut[p] * zfi * zfj * __builtin_amdgcn_rcpf(r);
        unsafeAtomicAdd(&out[i], contrib);
    }
}

extern "C" void kernel_launch(void* const* d_in, const int* in_sizes, int n_in,
                              void* d_out, int out_size, void* d_ws, size_t ws_size,
                              hipStream_t stream) {
    // setup_inputs order: N, Zf, rij, cutoff_values, idx_i, idx_j,
    //                     adiv, apow, c1, c2, c3, c4, a1, a2, a3, a4
    const float* Zf = (const float*)d_in[1];
    const float* rij = (const float*)d_in[2];
    const float* cutv = (const float*)d_in[3];
    const int* idx_i = (const int*)d_in[4];
    const int* idx_j = (const int*)d_in[5];
    const float* adiv = (const float*)d_in[6];
    const float* apow = (const float*)d_in[7];
    const float* c1 = (const float*)d_in[8];
    const float* c2 = (const float*)d_in[9];
    const float* c3 = (const float*)d_in[10];
    const float* c4 = (const float*)d_in[11];
    const float* a1 = (const float*)d_in[12];
    const float* a2 = (const float*)d_in[13];
    const float* a3 = (const float*)d_in[14];
    const float* a4 = (const float*)d_in[15];

    float* out = (float*)d_out;
    float* ws = (float*)d_ws;

    const int N = in_sizes[1];
    const int P = in_sizes[2];

    const size_t need = 64 + (size_t)N * sizeof(float2);
    const bool cached = ws_size >= need;

    zbl_consts_kernel<<<1, 32, 0, stream>>>(adiv, apow, c1, c2, c3, c4, a1, a2, a3, a4, ws);

    const int ablocks = (N + 255) / 256;
    zbl_atoms_kernel<<<ablocks, 256, 0, stream>>>(Zf, out, ws, N, cached ? 1 : 0);

    const int ntiles = P / TILE_PAIRS;
    if (ntiles > 0) {
        int blocks = (ntiles + WAVES_PER_BLOCK - 1) / WAVES_PER_BLOCK;
        if (blocks > 3072) blocks = 3072;   // ~24.5K waves; >=5 tiles/wave amortizes setup
        if (cached)
            zbl_pairs_kernel<true><<<blocks, 256, 0, stream>>>(Zf, rij, cutv, idx_i, idx_j, ws,
                                                               out, ntiles);
        else
            zbl_pairs_kernel<false><<<blocks, 256, 0, stream>>>(Zf, rij, cutv, idx_i, idx_j, ws,
                                                                out, ntiles);
    }
    const int start = ntiles * TILE_PAIRS;
    if (start < P) {
        zbl_pairs_tail_kernel<<<1, TILE_PAIRS, 0, stream>>>(Zf, rij, cutv, idx_i, idx_j, ws, out,
                                                            start, P, cached ? 1 : 0);
    }
}